// TopKRouter_90263032692930
// MI455X (gfx1250) — compile-verified
//
#include <hip/hip_runtime.h>
#include <hip/hip_bf16.h>
#include <stdint.h>

// ---------------- problem constants (match reference) ----------------
#define T_TOKENS 8192
#define HID      4096
#define NEXP     64
#define TOPK_K   8

// ---------------- tiling ----------------
#define BM        32                 // tokens per block (2 WMMA M tiles)
#define BK        64                 // K chunk staged in LDS per iteration
#define LDS_STR   (BK + 4)           // padded row stride (floats) to spread banks
#define NTHREADS  256                // 8 waves (wave32); one 16x16 tile per wave

typedef __attribute__((ext_vector_type(2))) float v2f;
typedef __attribute__((ext_vector_type(4))) float v4f;
typedef __attribute__((ext_vector_type(8))) float v8f;

// exact pointee type the async builtin wants: int __vector(4)
typedef int v4i_vs __attribute__((vector_size(16)));
typedef __attribute__((address_space(1))) v4i_vs* g_v4i_ptr;   // "__device__" in diags
typedef __attribute__((address_space(3))) v4i_vs* l_v4i_ptr;   // "__shared__" in diags

// ---------------- async global->LDS (CDNA5) with fallback ----------------
#if defined(__has_builtin)
#if __has_builtin(__builtin_amdgcn_global_load_async_to_lds_b128)
#define USE_ASYNC_LDS 1
#endif
#endif
#ifndef USE_ASYNC_LDS
#define USE_ASYNC_LDS 0
#endif

__device__ __forceinline__ void cp16_g2l(const float* gp, float* lp) {
#if USE_ASYNC_LDS
    g_v4i_ptr gv = reinterpret_cast<g_v4i_ptr>(reinterpret_cast<uintptr_t>(gp));
    l_v4i_ptr lv = reinterpret_cast<l_v4i_ptr>(reinterpret_cast<uintptr_t>(lp));
    __builtin_amdgcn_global_load_async_to_lds_b128(gv, lv, /*offset=*/0, /*cpol=*/0);
#else
    *reinterpret_cast<v4f*>(lp) = *reinterpret_cast<const v4f*>(gp);
#endif
}

__device__ __forceinline__ void async_wait0() {
#if USE_ASYNC_LDS
#if defined(__has_builtin) && __has_builtin(__builtin_amdgcn_s_wait_asynccnt)
    __builtin_amdgcn_s_wait_asynccnt(0);
#else
    asm volatile("s_wait_asynccnt 0" ::: "memory");
#endif
#endif
}

// stage one (x-tile, W-tile) K chunk into the given LDS buffers (async)
__device__ __forceinline__ void stage_chunk(const float* __restrict__ x,
                                            const float* __restrict__ w,
                                            float* __restrict__ xsb,
                                            float* __restrict__ wsb,
                                            int tok0, int k0, int tid) {
    // x tile: 32 rows x 64 floats = 512 x b128 -> 2 per thread
#pragma unroll
    for (int i = 0; i < (BM * BK / 4) / NTHREADS; ++i) {
        const int idx = i * NTHREADS + tid;
        const int r = idx >> 4;               // token row 0..31
        const int c = (idx & 15) << 2;        // float col 0..60 step 4
        cp16_g2l(&x[(size_t)(tok0 + r) * HID + k0 + c], &xsb[r * LDS_STR + c]);
    }
    // W tile: 64 rows x 64 floats = 1024 x b128 -> 4 per thread
#pragma unroll
    for (int i = 0; i < (NEXP * BK / 4) / NTHREADS; ++i) {
        const int idx = i * NTHREADS + tid;
        const int r = idx >> 4;               // expert row 0..63
        const int c = (idx & 15) << 2;
        cp16_g2l(&w[(size_t)r * HID + k0 + c], &wsb[r * LDS_STR + c]);
    }
}

// ---------------- kernel ----------------
__global__ __launch_bounds__(NTHREADS) void
router_topk_kernel(const float* __restrict__ x,     // [T_TOKENS, HID]
                   const float* __restrict__ w,     // [NEXP, HID]
                   float* __restrict__ probs,       // [T_TOKENS, NEXP]
                   float* __restrict__ rmap)        // [T_TOKENS, NEXP] (0/1)
{
    __shared__ float xs[2][BM   * LDS_STR];   // double-buffered x tile 32x64
    __shared__ float ws[2][NEXP * LDS_STR];   // double-buffered W tile 64x64
    __shared__ float lg[BM * NEXP];           // logits 32 x 64

    const int tid  = threadIdx.x;
    const int wave = tid >> 5;             // 0..7
    const int lane = tid & 31;
    const int half = lane >> 4;            // lane half selects K pair (A/B frag)
    const int l16  = lane & 15;            // M index (A) / N index (B)
    const int mt   = wave >> 2;            // M tile 0..1 (token group of 16)
    const int nt   = wave & 3;             // N tile 0..3 (expert group of 16)
    const int mb   = mt * 16;
    const int nb   = nt * 16;

    const int tok0 = blockIdx.x * BM;

    v8f acc = {};                          // 16x16 f32 accumulator (8 VGPRs)

    // prime the pipeline: stage chunk 0 into buffer 0
    stage_chunk(x, w, xs[0], ws[0], tok0, 0, tid);

    int buf = 0;
    for (int k0 = 0; k0 < HID; k0 += BK) {
        async_wait0();            // this wave's outstanding async copies done
        __syncthreads();          // all waves' copies done -> buf is ready

        // prefetch next chunk into the other buffer; DMA overlaps WMMA below
        if (k0 + BK < HID)
            stage_chunk(x, w, xs[buf ^ 1], ws[buf ^ 1], tok0, k0 + BK, tid);

        // ---- 16 WMMA steps over the K chunk (one 16x16 tile per wave) ----
        // A frag (16x4 f32): lane l -> row M=mb+l16, VGPRs = K {kk+2*half, +1}
        // B frag (4x16 f32): lane l -> col N=nb+l16, VGPRs = K {kk+2*half, +1}
        //   B[k][n] = W[expert n][k]  (logits = x @ W^T)
        const float* xrow = &xs[buf][(mb + l16) * LDS_STR + half * 2];
        const float* wrow = &ws[buf][(nb + l16) * LDS_STR + half * 2];
#pragma unroll
        for (int kk = 0; kk < BK; kk += 4) {
            v2f a = *reinterpret_cast<const v2f*>(xrow + kk);
            v2f b = *reinterpret_cast<const v2f*>(wrow + kk);
            acc = __builtin_amdgcn_wmma_f32_16x16x4_f32(
                /*neg_a=*/false, a, /*neg_b=*/false, b,
                /*c_mod=*/(short)0, acc, /*reuse_a=*/false, /*reuse_b=*/false);
        }
        __syncthreads();          // all waves done reading buf; next iter may refill it
        buf ^= 1;
    }

    // ---- spill 16x16 logits tiles to LDS ----
    // C/D layout: VGPR i, lanes0-15: M=i, N=lane; lanes16-31: M=i+8, N=lane-16
#pragma unroll
    for (int i = 0; i < 8; ++i) {
        const int m = mb + i + half * 8;
        lg[m * NEXP + nb + l16] = acc[i];
    }
    __syncthreads();

    // ---- per-token softmax + top-8 (softmax is monotonic: topk by logits) ----
    if (tid < BM) {
        const float* row = &lg[tid * NEXP];

        float mx = row[0];
#pragma unroll 4
        for (int e = 1; e < NEXP; ++e) mx = fmaxf(mx, row[e]);

        float denom = 0.0f;
#pragma unroll 4
        for (int e = 0; e < NEXP; ++e) denom += __expf(row[e] - mx);
        const float inv = 1.0f / denom;

        float* prow = probs + (size_t)(tok0 + tid) * NEXP;
        float* mrow = rmap  + (size_t)(tok0 + tid) * NEXP;
#pragma unroll 4
        for (int e = 0; e < NEXP; ++e) { prow[e] = 0.0f; mrow[e] = 0.0f; }

        unsigned long long taken = 0ull;
#pragma unroll
        for (int k = 0; k < TOPK_K; ++k) {
            int best = 0;
            float bv = -__builtin_inff();
            for (int e = 0; e < NEXP; ++e) {
                const bool used = (taken >> e) & 1ull;
                if (!used && row[e] > bv) { bv = row[e]; best = e; }  // ties -> lowest idx
            }
            taken |= (1ull << best);
            prow[best] = __expf(bv - mx) * inv;
            mrow[best] = 1.0f;
        }
    }
}

// ---------------- launcher ----------------
extern "C" void kernel_launch(void* const* d_in, const int* in_sizes, int n_in,
                              void* d_out, int out_size, void* d_ws, size_t ws_size,
                              hipStream_t stream) {
    (void)in_sizes; (void)n_in; (void)out_size; (void)d_ws; (void)ws_size;
    const float* x = (const float*)d_in[0];      // [8192, 4096] f32
    const float* w = (const float*)d_in[1];      // [64, 4096] f32
    float* probs = (float*)d_out;                                   // output 0
    float* rmap  = probs + (size_t)T_TOKENS * NEXP;                 // output 1 (bool as 0/1)

    dim3 grid(T_TOKENS / BM);   // 256 blocks
    dim3 block(NTHREADS);       // 8 waves of 32
    router_topk_kernel<<<grid, block, 0, stream>>>(x, w, probs, rmap);
}